// CrossAttention_26044681683354
// MI455X (gfx1250) — compile-verified
//
#include <hip/hip_runtime.h>
#include <math.h>

typedef float v2f __attribute__((ext_vector_type(2)));
typedef float v8f __attribute__((ext_vector_type(8)));

#define HH     64
#define WW     64
#define NTOK   4096
#define CEMB   256
#define NB     16
#define NHEADS 8
#define NPTS   4

// ---------------------------------------------------------------------------
// Positional encoding: pe[c, n] with c = 4*i + rem, n = h*64 + w
//   rem 0: sin((w+1)*div_i)  rem 1: cos((w+1)*div_i)
//   rem 2: sin((h+1)*div_i)  rem 3: cos((h+1)*div_i)
//   div_i = exp(2*i * (-ln(10000)/128))
// ---------------------------------------------------------------------------
__device__ __forceinline__ float pe_val(int c, int n) {
  const int i = c >> 2;
  const int rem = c & 3;
  const float dv = expf((float)(2 * i) * (-9.210340371976184f / 128.0f));
  const float coord = (rem < 2) ? (float)((n & 63) + 1) : (float)((n >> 6) + 1);
  const float arg = coord * dv;
  return (rem & 1) ? cosf(arg) : sinf(arg);
}

// q_flat[b][n][c] = query[b][c][n] + pe(c, n)   (32x32 LDS tile transpose)
__global__ __launch_bounds__(256) void prep_q_kernel(const float* __restrict__ query,
                                                     float* __restrict__ qflat) {
  __shared__ float tile[32][33];
  const int b = blockIdx.z;
  const int n0 = blockIdx.x * 32;
  const int c0 = blockIdx.y * 32;
  const int tx = threadIdx.x, ty = threadIdx.y;
  const float* qin = query + (size_t)b * CEMB * NTOK;
#pragma unroll
  for (int j = 0; j < 4; ++j) {
    const int c = c0 + ty + j * 8;
    const int n = n0 + tx;
    tile[ty + j * 8][tx] = qin[(size_t)c * NTOK + n] + pe_val(c, n);
  }
  __syncthreads();
  float* qo = qflat + (size_t)b * NTOK * CEMB;
#pragma unroll
  for (int j = 0; j < 4; ++j) {
    qo[(size_t)(n0 + ty + j * 8) * CEMB + (c0 + tx)] = tile[tx][ty + j * 8];
  }
}

// ---------------------------------------------------------------------------
// WMMA f32 GEMM:  out[m, c] = sum_k A[m,k] * Wt[c,k] + bias[c]   (per batch b)
//   A: row-major (M x K) or K-major (K x M, used for `value` directly)
//   MODE 0 (OUT_NM):        out (B, M, Cout)
//   MODE 1 (OUT_HEADSPLIT): out (B, HEADS, M, 32)       [for v_proj]
//   MODE 2 (OUT_FINAL):     out (B, C, M) transposed, += bias + 2*q_flat
// Workgroup = 8 waves; tile 64(M) x 64(N); K staged in LDS chunks of 32.
// Each wave: one 16-row strip, two 16x16 accumulators (n0, n0+16).
// ---------------------------------------------------------------------------
enum { OUT_NM = 0, OUT_HEADSPLIT = 1, OUT_FINAL = 2 };

template <int MODE, bool AKMAJ>
__global__ __launch_bounds__(256) void wmma_gemm_kernel(
    const float* __restrict__ A, const float* __restrict__ Wt,
    const float* __restrict__ bias, const float* __restrict__ qflat,
    float* __restrict__ out, int Cout) {
  constexpr int MT = 64, KC = 32, K = 256, M = NTOK;
  constexpr int LDA = 72;  // pad: float4-aligned, 2-row shift = 16 banks
  __shared__ float As[KC][LDA];
  __shared__ float Bs[KC][LDA];
  __shared__ float Tr[8][16][17];  // per-wave transpose staging (OUT_FINAL)

  const int b = blockIdx.z;
  const int m_blk = blockIdx.y * MT;
  const int n_blk = blockIdx.x * 64;
  const int tid = threadIdx.x;
  const int lane = tid & 31;
  const int wv = tid >> 5;
  const int wm = wv & 3, wn = wv >> 2;
  const int row = lane & 15, hi = lane >> 4;
  const int m0 = wm * 16;
  const int n0a = wn * 32, n0b = wn * 32 + 16;

  const float* Ab = A + (size_t)b * M * K;

  v8f acc0 = {};
  v8f acc1 = {};

  for (int kt = 0; kt < K; kt += KC) {
    if (AKMAJ) {  // A[k*M + m], coalesced along m
#pragma unroll
      for (int it = 0; it < 2; ++it) {
        const int id = tid + it * 256;
        const int kl = id >> 4;
        const int mc = (id & 15) * 4;
        const float4 vv = *(const float4*)(Ab + (size_t)(kt + kl) * M + m_blk + mc);
        *(float4*)&As[kl][mc] = vv;
      }
    } else {  // A[m*K + k], coalesced along k
#pragma unroll
      for (int it = 0; it < 2; ++it) {
        const int id = tid + it * 256;
        const int ml = id >> 3;
        const int kl = (id & 7) * 4;
        const float4 vv = *(const float4*)(Ab + (size_t)(m_blk + ml) * K + kt + kl);
        As[kl + 0][ml] = vv.x; As[kl + 1][ml] = vv.y;
        As[kl + 2][ml] = vv.z; As[kl + 3][ml] = vv.w;
      }
    }
    // Weights (Cout x K) -> Bs[k][n], zero-fill rows beyond Cout
#pragma unroll
    for (int it = 0; it < 2; ++it) {
      const int id = tid + it * 256;
      const int nl = id >> 3;
      const int kl = (id & 7) * 4;
      float4 vv = make_float4(0.f, 0.f, 0.f, 0.f);
      if (n_blk + nl < Cout)
        vv = *(const float4*)(Wt + (size_t)(n_blk + nl) * K + kt + kl);
      Bs[kl + 0][nl] = vv.x; Bs[kl + 1][nl] = vv.y;
      Bs[kl + 2][nl] = vv.z; Bs[kl + 3][nl] = vv.w;
    }
    __syncthreads();
#pragma unroll
    for (int k0 = 0; k0 < KC; k0 += 4) {
      // 16x16x4 f32 fragments per ISA layout:
      //   A: lane(m=row), vgpr j -> K = k0 + hi*2 + j
      //   B: lane(n=row), vgpr j -> K = k0 + hi*2 + j
      const int kk = k0 + hi * 2;
      v2f a, fb0, fb1;
      a[0] = As[kk][m0 + row];      a[1] = As[kk + 1][m0 + row];
      fb0[0] = Bs[kk][n0a + row];   fb0[1] = Bs[kk + 1][n0a + row];
      fb1[0] = Bs[kk][n0b + row];   fb1[1] = Bs[kk + 1][n0b + row];
      acc0 = __builtin_amdgcn_wmma_f32_16x16x4_f32(false, a, false, fb0,
                                                   (short)0, acc0, false, false);
      acc1 = __builtin_amdgcn_wmma_f32_16x16x4_f32(false, a, false, fb1,
                                                   (short)0, acc1, false, false);
    }
    __syncthreads();
  }

  // Epilogue. C/D fragment: element (m, n) at vgpr r, lane L: n = L%16, m = r + (L>=16)*8
#pragma unroll
  for (int half = 0; half < 2; ++half) {
    const v8f acc = half ? acc1 : acc0;
    const int n0 = half ? n0b : n0a;
    const int cg = n_blk + n0 + row;
    if (MODE == OUT_NM) {
      if (cg < Cout) {
        const float bv = bias[cg];
#pragma unroll
        for (int r = 0; r < 8; ++r) {
          const int mg = m_blk + m0 + r + hi * 8;
          out[((size_t)b * M + mg) * Cout + cg] = acc[r] + bv;
        }
      }
    } else if (MODE == OUT_HEADSPLIT) {
      const float bv = bias[cg];
      const int head = cg >> 5, dd = cg & 31;
#pragma unroll
      for (int r = 0; r < 8; ++r) {
        const int mg = m_blk + m0 + r + hi * 8;
        out[(((size_t)b * NHEADS + head) * M + mg) * 32 + dd] = acc[r] + bv;
      }
    } else {  // OUT_FINAL: val = acc + b_out[c] + 2*qflat[b,m,c]; store (B,C,N)
      const float bv = bias[cg];
      const float* qf = qflat + ((size_t)b * M + m_blk + m0) * CEMB + cg;
#pragma unroll
      for (int r = 0; r < 8; ++r) {
        const int ml = r + hi * 8;
        Tr[wv][ml][row] = acc[r] + bv + 2.0f * qf[(size_t)ml * CEMB];
      }
      // in-wave LDS transpose (DS ops are in-order within a wave)
#pragma unroll
      for (int r = 0; r < 8; ++r) {
        const int cl = r + hi * 8;
        out[((size_t)b * CEMB + (n_blk + n0 + cl)) * NTOK + (m_blk + m0 + row)] =
            Tr[wv][row][cl];
      }
    }
  }
}

// ---------------------------------------------------------------------------
// Deformable sampling: one wave per token, lane = channel within head (0..31).
// vproj layout (B, HEADS, N, 32) -> each corner gather is a contiguous 128B row.
// ---------------------------------------------------------------------------
__global__ __launch_bounds__(256) void sample_kernel(const float* __restrict__ vproj,
                                                     const float* __restrict__ offs,
                                                     const float* __restrict__ logits,
                                                     float* __restrict__ samp) {
  const int wave = (int)((blockIdx.x * (size_t)blockDim.x + threadIdx.x) >> 5);
  const int lane = threadIdx.x & 31;
  const int b = wave >> 12;       // / NTOK
  const int n = wave & (NTOK - 1);
  // reference_points: ref[n] = ((n%64)/63, (n/64)/63)
  const float ref0 = (float)(n & 63) * (1.0f / 63.0f);
  const float ref1 = (float)(n >> 6) * (1.0f / 63.0f);
  const float* offp = offs + ((size_t)b * NTOK + n) * (NHEADS * NPTS * 2);
  const float* lg = logits + ((size_t)b * NTOK + n) * (NHEADS * NPTS);
  float* outp = samp + ((size_t)b * NTOK + n) * CEMB;

#pragma unroll
  for (int head = 0; head < NHEADS; ++head) {
    // softmax over the 4 points
    float e[NPTS];
    float mx = -3.4e38f;
#pragma unroll
    for (int p = 0; p < NPTS; ++p) { e[p] = lg[head * NPTS + p]; mx = fmaxf(mx, e[p]); }
    float den = 0.f;
#pragma unroll
    for (int p = 0; p < NPTS; ++p) { e[p] = __expf(e[p] - mx); den += e[p]; }
    const float inv = 1.0f / den;

    const float* vb = vproj + (((size_t)b * NHEADS + head) * NTOK) * 32;
    float acc = 0.f;
#pragma unroll
    for (int p = 0; p < NPTS; ++p) {
      const float ox = offp[(head * NPTS + p) * 2 + 0];
      const float oy = offp[(head * NPTS + p) * 2 + 1];
      const float ix = (ref0 + ox * (1.0f / (float)WW)) * (float)WW - 0.5f;
      const float iy = (ref1 + oy * (1.0f / (float)HH)) * (float)HH - 0.5f;
      const float x0f = floorf(ix), y0f = floorf(iy);
      const float wx = ix - x0f, wy = iy - y0f;
      const int x0 = (int)x0f, y0 = (int)y0f;

      float s = 0.f;
#pragma unroll
      for (int cy = 0; cy < 2; ++cy) {
#pragma unroll
        for (int cx = 0; cx < 2; ++cx) {
          const int xx = x0 + cx, yy = y0 + cy;
          const bool valid = (xx >= 0) & (xx < WW) & (yy >= 0) & (yy < HH);
          const int xc = min(max(xx, 0), WW - 1);
          const int yc = min(max(yy, 0), HH - 1);
          const float wgt = (cx ? wx : 1.0f - wx) * (cy ? wy : 1.0f - wy);
          const float v = vb[((size_t)(yc * WW + xc)) * 32 + lane];
          s += valid ? wgt * v : 0.0f;
        }
      }
      acc += (e[p] * inv) * s;
    }
    outp[head * 32 + lane] = acc;
  }
}

// ---------------------------------------------------------------------------
extern "C" void kernel_launch(void* const* d_in, const int* in_sizes, int n_in,
                              void* d_out, int out_size, void* d_ws, size_t ws_size,
                              hipStream_t stream) {
  const float* query  = (const float*)d_in[0];
  const float* value  = (const float*)d_in[1];
  const float* W_off  = (const float*)d_in[2];
  const float* b_off  = (const float*)d_in[3];
  const float* W_attn = (const float*)d_in[4];
  const float* b_attn = (const float*)d_in[5];
  const float* W_val  = (const float*)d_in[6];
  const float* b_val  = (const float*)d_in[7];
  const float* W_out  = (const float*)d_in[8];
  const float* b_out  = (const float*)d_in[9];
  float* out = (float*)d_out;

  char* ws = (char*)d_ws;
  const size_t szBNC = (size_t)NB * NTOK * CEMB * sizeof(float);  // 64 MB
  float* qflat = (float*)ws;  ws += szBNC;
  float* vproj = (float*)ws;  ws += szBNC;
  float* offb  = (float*)ws;  ws += (size_t)NB * NTOK * 64 * sizeof(float);
  float* attnl = (float*)ws;  ws += (size_t)NB * NTOK * 32 * sizeof(float);
  float* samp  = (float*)ws;

  // 1) q_flat = transpose(query) + PE
  prep_q_kernel<<<dim3(NTOK / 32, CEMB / 32, NB), dim3(32, 8), 0, stream>>>(query, qflat);
  // 2) v_proj = value^T @ W_val^T + b_val  (A consumed K-major, head-split out)
  wmma_gemm_kernel<OUT_HEADSPLIT, true>
      <<<dim3(CEMB / 64, NTOK / 64, NB), 256, 0, stream>>>(value, W_val, b_val,
                                                           nullptr, vproj, CEMB);
  // 3) offsets = q_flat @ W_off^T + b_off
  wmma_gemm_kernel<OUT_NM, false>
      <<<dim3(1, NTOK / 64, NB), 256, 0, stream>>>(qflat, W_off, b_off,
                                                   nullptr, offb, 64);
  // 4) attn logits = q_flat @ W_attn^T + b_attn
  wmma_gemm_kernel<OUT_NM, false>
      <<<dim3(1, NTOK / 64, NB), 256, 0, stream>>>(qflat, W_attn, b_attn,
                                                   nullptr, attnl, 32);
  // 5) softmax + bilinear deformable sampling -> (B, N, C)
  sample_kernel<<<dim3(NB * NTOK / 8), 256, 0, stream>>>(vproj, offb, attnl, samp);
  // 6) out = samp @ W_out^T + b_out + 2*q_flat, stored transposed (B, C, H, W)
  wmma_gemm_kernel<OUT_FINAL, false>
      <<<dim3(CEMB / 64, NTOK / 64, NB), 256, 0, stream>>>(samp, W_out, b_out,
                                                           qflat, out, CEMB);
}